// SIFT_42949673316
// MI455X (gfx1250) — compile-verified
//
#include <hip/hip_runtime.h>
#include <math.h>

// SIFT descriptor for 65x65 patches on gfx1250 (MI455X).
// One block (160 threads = 5 wave32) per patch-channel; 3072 blocks.
// y-direction pooling done with V_WMMA_F32_16X16X4_F32 (wave 0).
// Gaussian /S^2 normalization dropped: uniform scale cancels in L2 normalize.

#define PS        65
#define TILE_W    66          // padded row stride for patch tile
#define HX_ROWS   68          // 65 rows padded to K-multiple of 4 (zeroed tail)
#define HX_STRIDE 33          // 32 cols (8 angle * 4 px) + 1 pad
#define NPIX      (PS * PS)
#define NTHREADS  160

typedef __attribute__((ext_vector_type(2))) float v2f;
typedef __attribute__((ext_vector_type(8))) float v8f;

// triangular pooling kernel value, k in [0,25]: (13 - |k + 0.5 - 13|)/13
__device__ __forceinline__ float tri_w(float k) {
    return (13.0f - fabsf(k + 0.5f - 13.0f)) * (1.0f / 13.0f);
}

// atan2 via min/max ratio + degree-11 odd minimax poly (~1e-5 rad abs error).
__device__ __forceinline__ float fast_atan2(float y, float x) {
    float ax = fabsf(x), ay = fabsf(y);
    float mx = fmaxf(ax, ay);
    float mn = fminf(ax, ay);
    float q  = mn * __builtin_amdgcn_rcpf(fmaxf(mx, 1e-30f));  // [0,1]
    float t  = q * q;
    float p  = fmaf(t, -0.01172120f, 0.05265332f);
    p = fmaf(t, p, -0.11643287f);
    p = fmaf(t, p,  0.19354346f);
    p = fmaf(t, p, -0.33262347f);
    p = fmaf(t, p,  0.99997726f);
    float r = q * p;
    r = (ay > ax)   ? (1.57079632679489662f - r) : r;
    r = (x < 0.0f)  ? (3.14159265358979323f - r) : r;
    return copysignf(r, y);
}

__global__ __launch_bounds__(NTHREADS)
void sift_patch_kernel(const float* __restrict__ img, float* __restrict__ out)
{
    __shared__ float tile[PS * TILE_W];            // staged patch
    __shared__ float hx[HX_ROWS * HX_STRIDE];      // x-pooled field [y][a*4+px]
    __shared__ float dbuf[128];                    // pooled descriptor (pre-norm)

    const int f   = blockIdx.x;                    // 0..3071
    const int tid = threadIdx.x;                   // 0..159

    // f = ((b*3+c)*8 + ih)*8 + iw ; chan = b*3+c = f>>6
    const int iw   = f & 7;
    const int ih   = (f >> 3) & 7;
    const int chan = f >> 6;
    const size_t base = ((size_t)chan * 520 + (size_t)ih * 65) * 520 + (size_t)iw * 65;

    // ---------- Phase A: stage patch into LDS, zero hx ----------
    for (int i = tid; i < NPIX; i += NTHREADS) {
        int y = i / PS, x = i - y * PS;
        tile[y * TILE_W + x] = img[base + (size_t)y * 520 + x];
    }
    for (int i = tid; i < HX_ROWS * HX_STRIDE; i += NTHREADS) hx[i] = 0.0f;
    __syncthreads();

    // ---------- Phase B: gradients, orientation, x-direction pooling ----------
    // thread t < 130 owns half of row y = t>>1 (cols [0,33) or [33,65))
    if (tid < 130) {
        const int y  = tid >> 1;
        const int x0 = (tid & 1) ? 33 : 0;
        const int xe = (tid & 1) ? 65 : 33;
        const float* rowc = &tile[y * TILE_W];
        const float* rowm = &tile[(y > 0 ? y - 1 : 0) * TILE_W];
        const float* rowq = &tile[(y < PS - 1 ? y + 1 : PS - 1) * TILE_W];
        float* hrow = &hx[y * HX_STRIDE];
        const float dyf = (float)(y - 32);
        const float dy2 = dyf * dyf;

        float xm  = rowc[x0 > 0 ? x0 - 1 : 0];
        float cur = rowc[x0];
        for (int x = x0; x < xe; ++x) {
            float xp = rowc[x < PS - 1 ? x + 1 : PS - 1];
            float ym = rowm[x];
            float yp = rowq[x];
            float gx = 0.5f * (xp - xm);
            float gy = 0.5f * (yp - ym);
            xm = cur; cur = xp;

            float g2  = fmaf(gx, gx, fmaf(gy, gy, 1e-10f));
            float mag = __builtin_amdgcn_sqrtf(g2);
            float dxf = (float)(x - 32);
            float w   = __expf(fmaf(dxf, dxf, dy2) * -2.36686390532544378698e-4f);
            float m   = mag * w;

            // o = (8/2pi)*(atan2+2pi) = atan2*1.2732... + 8, in (4,12]
            float ang  = fast_atan2(gy, gx + 1e-10f);
            float o    = fmaf(ang, 1.27323954473516268615f, 8.0f);
            float bo0f = floorf(o);
            float wo1  = o - bo0f;
            int   b0   = ((int)bo0f) & 7;
            int   b1   = (b0 + 1) & 7;
            float m0   = (1.0f - wo1) * m;
            float m1   = wo1 * m;

            // x pooling: padded coord u = x+6; window p covers u in [16p,16p+25]
            int   pA = (x + 6) >> 4;
            float kA = (float)(x + 6 - (pA << 4));     // in [0,15]
            if (pA <= 3) {
                float wxA = tri_w(kA);
                atomicAdd(hrow + b0 * 4 + pA, m0 * wxA);
                atomicAdd(hrow + b1 * 4 + pA, m1 * wxA);
            }
            int pB = pA - 1;
            if (pB >= 0 && kA <= 9.0f) {               // kB = kA+16 <= 25
                float wxB = tri_w(kA + 16.0f);
                atomicAdd(hrow + b0 * 4 + pB, m0 * wxB);
                atomicAdd(hrow + b1 * 4 + pB, m1 * wxB);
            }
        }
    }
    __syncthreads();

    // ---------- Phase C: y-pooling GEMM pooled[4x32] = Wy[4x68] * hx[68x32]
    // via V_WMMA_F32_16X16X4_F32, wave 0 only (EXEC all-ones within wave).
    if (tid < 32) {
        const int lane = tid;
        const int m16  = lane & 15;                // A: M row (py if <4); B: N low bits
        const int kOff = (lane >> 4) << 1;         // K pair handled by this lane half
        v8f c0 = {0.f,0.f,0.f,0.f,0.f,0.f,0.f,0.f};
        v8f c1 = {0.f,0.f,0.f,0.f,0.f,0.f,0.f,0.f};
#pragma unroll 1
        for (int ks = 0; ks < 17; ++ks) {
            int ya = ks * 4 + kOff, yb = ya + 1;
            // A[M][K] = Wy[py=m16][y]: triangular weight of row y for bin py
            v2f a; a.x = 0.0f; a.y = 0.0f;
            if (m16 <= 3) {
                int ka = ya - 16 * m16 + 6;
                int kb = ka + 1;
                if (ka >= 0 && ka <= 25) a.x = tri_w((float)ka);
                if (kb >= 0 && kb <= 25) a.y = tri_w((float)kb);
            }
            // B[K][N] rows from LDS (rows 65..67 are zero padding)
            v2f b0v, b1v;
            b0v.x = hx[ya * HX_STRIDE + m16];
            b0v.y = hx[yb * HX_STRIDE + m16];
            b1v.x = hx[ya * HX_STRIDE + 16 + m16];
            b1v.y = hx[yb * HX_STRIDE + 16 + m16];
            c0 = __builtin_amdgcn_wmma_f32_16x16x4_f32(false, a, false, b0v,
                                                       (short)0, c0, false, false);
            c1 = __builtin_amdgcn_wmma_f32_16x16x4_f32(false, a, false, b1v,
                                                       (short)0, c1, false, false);
        }
        // D layout (16x16 f32): lane<16, VGPR r -> M=r (=py), N=lane. Rows 4..15 zero.
        if (lane < 16) {
#pragma unroll
            for (int r = 0; r < 4; ++r) {
                int n0 = lane;        // tile 0: cols 0..15  -> a = n>>2, px = n&3
                dbuf[(n0 >> 2) * 16 + r * 4 + (n0 & 3)] = c0[r];
                int n1 = lane + 16;   // tile 1: cols 16..31
                dbuf[(n1 >> 2) * 16 + r * 4 + (n1 & 3)] = c1[r];
            }
        }
    }
    __syncthreads();

    // ---------- Phase D: L2 normalize -> clip(0.2) -> L2 normalize, store ----------
    if (tid < 32) {
        float v0 = dbuf[tid * 4 + 0];
        float v1 = dbuf[tid * 4 + 1];
        float v2 = dbuf[tid * 4 + 2];
        float v3 = dbuf[tid * 4 + 3];
        float ss = v0 * v0 + v1 * v1 + v2 * v2 + v3 * v3;
#pragma unroll
        for (int off = 16; off > 0; off >>= 1) ss += __shfl_xor(ss, off, 32);
        float r1 = 1.0f / fmaxf(sqrtf(ss), 1e-12f);
        v0 = fminf(fmaxf(v0 * r1, 0.0f), 0.2f);
        v1 = fminf(fmaxf(v1 * r1, 0.0f), 0.2f);
        v2 = fminf(fmaxf(v2 * r1, 0.0f), 0.2f);
        v3 = fminf(fmaxf(v3 * r1, 0.0f), 0.2f);
        float ss2 = v0 * v0 + v1 * v1 + v2 * v2 + v3 * v3;
#pragma unroll
        for (int off = 16; off > 0; off >>= 1) ss2 += __shfl_xor(ss2, off, 32);
        float r2 = 1.0f / fmaxf(sqrtf(ss2), 1e-12f);
        float* dst = out + (size_t)f * 128 + tid * 4;
        dst[0] = v0 * r2;
        dst[1] = v1 * r2;
        dst[2] = v2 * r2;
        dst[3] = v3 * r2;
    }
}

extern "C" void kernel_launch(void* const* d_in, const int* in_sizes, int n_in,
                              void* d_out, int out_size, void* d_ws, size_t ws_size,
                              hipStream_t stream) {
    (void)in_sizes; (void)n_in; (void)d_ws; (void)ws_size; (void)out_size;
    const float* img = (const float*)d_in[0];   // [16,3,520,520] f32
    float* out = (float*)d_out;                 // [3072,128] f32
    sift_patch_kernel<<<3072, NTHREADS, 0, stream>>>(img, out);
}